// AttentionLayer_38397007626388
// MI455X (gfx1250) — compile-verified
//
#include <hip/hip_runtime.h>

typedef float v2f __attribute__((ext_vector_type(2)));
typedef float v8f __attribute__((ext_vector_type(8)));

#define N_F   50000
#define N_R   25000
#define E_FF  400000
#define E_RR  200000
#define E_RF  400000
#define D_NODE 128
#define D_EDGE 64

__device__ __forceinline__ float lrelu(float x) { return x >= 0.0f ? x : 0.01f * x; }

__device__ __forceinline__ float wave_sum32(float v) {
#pragma unroll
  for (int off = 16; off > 0; off >>= 1) v += __shfl_xor(v, off, 32);
  return v;
}

// ---------------------------------------------------------------------------
// Generic fp32 GEMM: C[M,N] = A[M,K] @ B[K,N], row-major, via V_WMMA_F32_16X16X4_F32.
// One 32-lane wave per 16x16 output tile. Loads clamp rows (EXEC stays all-1s
// for WMMA); stores are predicated for the M-tail.
// A frag (16x4 f32): lanes 0-15 rows, v0/v1 = K0/K1; lanes 16-31 = K2/K3.
// B frag (4x16 f32): lanes = cols, halves split K the same way.
// C frag: VGPR v, lanes 0-15 -> row v, lanes 16-31 -> row v+8.
// ---------------------------------------------------------------------------
__global__ void gemm_f32_wmma(const float* __restrict__ A, const float* __restrict__ B,
                              float* __restrict__ C, int M, int N, int K) {
  const int lane = threadIdx.x;
  const int half = lane >> 4;
  const int l15  = lane & 15;
  const int rowBase = blockIdx.x * 16;
  const int col     = blockIdx.y * 16 + l15;

  int rowA = rowBase + l15;
  if (rowA > M - 1) rowA = M - 1;           // clamp: keep EXEC full for WMMA
  const float* Arow = A + (size_t)rowA * K;

  v8f acc = {};
  for (int k = 0; k < K; k += 4) {
    const int kb = k + half * 2;
    v2f a, b;
    a.x = Arow[kb];
    a.y = Arow[kb + 1];
    b.x = B[(size_t)kb * N + col];
    b.y = B[(size_t)(kb + 1) * N + col];
    acc = __builtin_amdgcn_wmma_f32_16x16x4_f32(false, a, false, b, (short)0, acc,
                                                false, false);
  }
#pragma unroll
  for (int v = 0; v < 8; ++v) {
    const int r = rowBase + v + half * 8;
    if (r < M) C[(size_t)r * N + col] = acc[v];
  }
}

// ---------------------------------------------------------------------------
// Final node update: Out[M,N] = Base + lrelu(A1@B1 (+ A2@B2)), K fixed per call.
// Same WMMA tiling; fuses the residual + leaky-ReLU epilogue.
// ---------------------------------------------------------------------------
__global__ void node_update_wmma(const float* __restrict__ A1, const float* __restrict__ B1,
                                 const float* __restrict__ A2, const float* __restrict__ B2,
                                 const float* __restrict__ Base, float* __restrict__ Out,
                                 int M, int N, int K) {
  const int lane = threadIdx.x;
  const int half = lane >> 4;
  const int l15  = lane & 15;
  const int rowBase = blockIdx.x * 16;
  const int col     = blockIdx.y * 16 + l15;

  int rowA = rowBase + l15;
  if (rowA > M - 1) rowA = M - 1;

  v8f acc = {};
  {
    const float* Arow = A1 + (size_t)rowA * K;
    for (int k = 0; k < K; k += 4) {
      const int kb = k + half * 2;
      v2f a, b;
      a.x = Arow[kb];
      a.y = Arow[kb + 1];
      b.x = B1[(size_t)kb * N + col];
      b.y = B1[(size_t)(kb + 1) * N + col];
      acc = __builtin_amdgcn_wmma_f32_16x16x4_f32(false, a, false, b, (short)0, acc,
                                                  false, false);
    }
  }
  if (A2 != nullptr) {                       // uniform branch: EXEC unchanged
    const float* Arow = A2 + (size_t)rowA * K;
    for (int k = 0; k < K; k += 4) {
      const int kb = k + half * 2;
      v2f a, b;
      a.x = Arow[kb];
      a.y = Arow[kb + 1];
      b.x = B2[(size_t)kb * N + col];
      b.y = B2[(size_t)(kb + 1) * N + col];
      acc = __builtin_amdgcn_wmma_f32_16x16x4_f32(false, a, false, b, (short)0, acc,
                                                  false, false);
    }
  }
#pragma unroll
  for (int v = 0; v < 8; ++v) {
    const int r = rowBase + v + half * 8;
    if (r < M) {
      const size_t idx = (size_t)r * N + col;
      Out[idx] = Base[idx] + lrelu(acc[v]);
    }
  }
}

// ---------------------------------------------------------------------------
// Per-node attention dots: d_k[i] = H[i,:] . a_k  for three 128-dim vectors.
// One wave per node, float4 per lane (32*4 = 128).
// ---------------------------------------------------------------------------
__global__ void node_dots(const float* __restrict__ H,
                          const float* __restrict__ a0, const float* __restrict__ a1,
                          const float* __restrict__ a2,
                          float* __restrict__ d0, float* __restrict__ d1,
                          float* __restrict__ d2, int n_nodes) {
  const int node = blockIdx.x * (blockDim.x >> 5) + (threadIdx.x >> 5);
  const int lane = threadIdx.x & 31;
  if (node >= n_nodes) return;
  const float4 hv = reinterpret_cast<const float4*>(H + (size_t)node * D_NODE)[lane];
  const float4 w0 = reinterpret_cast<const float4*>(a0)[lane];
  const float4 w1 = reinterpret_cast<const float4*>(a1)[lane];
  const float4 w2 = reinterpret_cast<const float4*>(a2)[lane];
  float p0 = hv.x * w0.x + hv.y * w0.y + hv.z * w0.z + hv.w * w0.w;
  float p1 = hv.x * w1.x + hv.y * w1.y + hv.z * w1.z + hv.w * w1.w;
  float p2 = hv.x * w2.x + hv.y * w2.y + hv.z * w2.z + hv.w * w2.w;
  p0 = wave_sum32(p0);
  p1 = wave_sum32(p1);
  p2 = wave_sum32(p2);
  if (lane == 0) { d0[node] = p0; d1[node] = p1; d2[node] = p2; }
}

// ---------------------------------------------------------------------------
// Edge scores: s = lrelu(dsrc[src] + ddst[dst] + Eproj[e,:].a_e); atomic segment
// sum of s at the dst node. One wave per edge, float2 per lane (32*2 = 64).
// ---------------------------------------------------------------------------
__global__ void edge_scores(const float* __restrict__ Eproj, const float* __restrict__ a_e,
                            const int* __restrict__ src, const int* __restrict__ dst,
                            const float* __restrict__ dsrc, const float* __restrict__ ddst,
                            float* __restrict__ s, float* __restrict__ sum_dst, int n_edges) {
  const int e = blockIdx.x * (blockDim.x >> 5) + (threadIdx.x >> 5);
  const int lane = threadIdx.x & 31;
  if (e >= n_edges) return;
  const float2 ev = reinterpret_cast<const float2*>(Eproj + (size_t)e * D_EDGE)[lane];
  const float2 av = reinterpret_cast<const float2*>(a_e)[lane];
  float p = ev.x * av.x + ev.y * av.y;
  p = wave_sum32(p);
  if (lane == 0) {
    const float sc = lrelu(dsrc[src[e]] + ddst[dst[e]] + p);
    s[e] = sc;
    atomicAdd(sum_dst + dst[e], sc);
  }
}

// ---------------------------------------------------------------------------
// Edge update: new_e = (s[e]/sum_src[src]) * (Hc[dst,:] + Eproj[e,:]).
// Writes new_e in place over Eproj (its d_out slot) and scatter-adds it into
// the dst aggregation buffer. One wave per edge, float2 per lane.
// ---------------------------------------------------------------------------
__global__ void edge_update(float* __restrict__ Eio, const float* __restrict__ Hc,
                            const int* __restrict__ src, const int* __restrict__ dst,
                            const float* __restrict__ s, const float* __restrict__ sum_src,
                            float* __restrict__ Agg, int n_edges) {
  const int e = blockIdx.x * (blockDim.x >> 5) + (threadIdx.x >> 5);
  const int lane = threadIdx.x & 31;
  if (e >= n_edges) return;
  const int sn = src[e], dn = dst[e];
  const float coef = s[e] / sum_src[sn];
  float2* erow = reinterpret_cast<float2*>(Eio + (size_t)e * D_EDGE);
  const float2 ev = erow[lane];
  const float2 hv = reinterpret_cast<const float2*>(Hc + (size_t)dn * D_EDGE)[lane];
  const float2 nv = { coef * (hv.x + ev.x), coef * (hv.y + ev.y) };
  erow[lane] = nv;
  float* ag = Agg + (size_t)dn * D_EDGE + 2 * lane;
  atomicAdd(ag + 0, nv.x);
  atomicAdd(ag + 1, nv.y);
}

__global__ void fill_zero(float* __restrict__ p, size_t n) {
  size_t i = (size_t)blockIdx.x * blockDim.x + threadIdx.x;
  const size_t stride = (size_t)gridDim.x * blockDim.x;
  for (; i < n; i += stride) p[i] = 0.0f;
}

// ---------------------------------------------------------------------------

extern "C" void kernel_launch(void* const* d_in, const int* in_sizes, int n_in,
                              void* d_out, int out_size, void* d_ws, size_t ws_size,
                              hipStream_t stream) {
  (void)in_sizes; (void)n_in; (void)out_size; (void)ws_size;

  const float* f_feat   = (const float*)d_in[0];
  const float* r_feat   = (const float*)d_in[1];
  const float* ff_efeat = (const float*)d_in[2];
  const float* rr_efeat = (const float*)d_in[3];
  const float* rf_efeat = (const float*)d_in[4];
  const float* Wf       = (const float*)d_in[5];
  const float* Wr       = (const float*)d_in[6];
  const float* We_ff    = (const float*)d_in[7];
  const float* We_rr    = (const float*)d_in[8];
  const float* We_rf    = (const float*)d_in[9];
  const float* a_ff     = (const float*)d_in[10];
  const float* a_rr     = (const float*)d_in[11];
  const float* a_rf     = (const float*)d_in[12];
  const float* Wec_ff   = (const float*)d_in[13];
  const float* Wec_rr   = (const float*)d_in[14];
  const float* Wec_rf   = (const float*)d_in[15];
  const float* Wnc_ff   = (const float*)d_in[16];
  const float* Wnc_rr   = (const float*)d_in[17];
  const float* Wnc_rf   = (const float*)d_in[18];
  const int* ff_src = (const int*)d_in[19];
  const int* ff_dst = (const int*)d_in[20];
  const int* rr_src = (const int*)d_in[21];
  const int* rr_dst = (const int*)d_in[22];
  const int* rf_src = (const int*)d_in[23];
  const int* rf_dst = (const int*)d_in[24];

  // d_out layout: new_f | new_r | new_eff | new_err | new_erf
  float* out_f   = (float*)d_out;
  float* out_r   = out_f + (size_t)N_F * D_NODE;
  float* out_eff = out_r + (size_t)N_R * D_NODE;
  float* out_err = out_eff + (size_t)E_FF * D_EDGE;
  float* out_erf = out_err + (size_t)E_RR * D_EDGE;

  // Workspace layout (floats)
  float* ws = (float*)d_ws;
  size_t off = 0;
  float* hf     = ws + off; off += (size_t)N_F * D_NODE;   // 6.4M
  float* hr     = ws + off; off += (size_t)N_R * D_NODE;   // 3.2M
  float* hfc_ff = ws + off; off += (size_t)N_F * D_EDGE;   // hf @ Wec_ff
  float* hfc_rf = ws + off; off += (size_t)N_F * D_EDGE;   // hf @ Wec_rf
  float* hrc_rr = ws + off; off += (size_t)N_R * D_EDGE;   // hr @ Wec_rr
  float* df0 = ws + off; off += N_F;    // hf . a_ff[0:128]
  float* df1 = ws + off; off += N_F;    // hf . a_ff[128:256]
  float* df2 = ws + off; off += N_F;    // hf . a_rf[128:256]
  float* dr0 = ws + off; off += N_R;    // hr . a_rr[0:128]
  float* dr1 = ws + off; off += N_R;    // hr . a_rr[128:256]
  float* dr2 = ws + off; off += N_R;    // hr . a_rf[0:128]
  float* s_ff = ws + off; off += E_FF;
  float* s_rr = ws + off; off += E_RR;
  float* s_rf = ws + off; off += E_RF;
  // zero-initialized region (contiguous, one fill)
  float* zero_base = ws + off;
  float* sum_f  = ws + off; off += N_F;
  float* sum_r  = ws + off; off += N_R;
  float* agg_ff = ws + off; off += (size_t)N_F * D_EDGE;
  float* agg_rf = ws + off; off += (size_t)N_F * D_EDGE;
  float* agg_rr = ws + off; off += (size_t)N_R * D_EDGE;
  const size_t n_zero = (size_t)(ws + off - zero_base);

  // 0) zero the accumulators
  fill_zero<<<dim3(2048), dim3(256), 0, stream>>>(zero_base, n_zero);

  const dim3 wave(32);
  #define TILES(m) (unsigned)(((m) + 15) / 16)

  // 1) node projections
  gemm_f32_wmma<<<dim3(TILES(N_F), D_NODE / 16), wave, 0, stream>>>(f_feat, Wf, hf, N_F, D_NODE, D_NODE);
  gemm_f32_wmma<<<dim3(TILES(N_R), D_NODE / 16), wave, 0, stream>>>(r_feat, Wr, hr, N_R, D_NODE, D_NODE);

  // 2) edge projections (straight into their d_out slots; updated in place later)
  gemm_f32_wmma<<<dim3(TILES(E_FF), D_EDGE / 16), wave, 0, stream>>>(ff_efeat, We_ff, out_eff, E_FF, D_EDGE, D_EDGE);
  gemm_f32_wmma<<<dim3(TILES(E_RR), D_EDGE / 16), wave, 0, stream>>>(rr_efeat, We_rr, out_err, E_RR, D_EDGE, D_EDGE);
  gemm_f32_wmma<<<dim3(TILES(E_RF), D_EDGE / 16), wave, 0, stream>>>(rf_efeat, We_rf, out_erf, E_RF, D_EDGE, D_EDGE);

  // 3) node -> edge converters
  gemm_f32_wmma<<<dim3(TILES(N_F), D_EDGE / 16), wave, 0, stream>>>(hf, Wec_ff, hfc_ff, N_F, D_EDGE, D_NODE);
  gemm_f32_wmma<<<dim3(TILES(N_F), D_EDGE / 16), wave, 0, stream>>>(hf, Wec_rf, hfc_rf, N_F, D_EDGE, D_NODE);
  gemm_f32_wmma<<<dim3(TILES(N_R), D_EDGE / 16), wave, 0, stream>>>(hr, Wec_rr, hrc_rr, N_R, D_EDGE, D_NODE);

  // 4) per-node attention dots (wave per node; 8 nodes per 256-thread block)
  node_dots<<<dim3((N_F + 7) / 8), dim3(256), 0, stream>>>(hf, a_ff, a_ff + 128, a_rf + 128,
                                                           df0, df1, df2, N_F);
  node_dots<<<dim3((N_R + 7) / 8), dim3(256), 0, stream>>>(hr, a_rr, a_rr + 128, a_rf,
                                                           dr0, dr1, dr2, N_R);

  // 5) edge scores + dst-segment score sums
  edge_scores<<<dim3((E_FF + 7) / 8), dim3(256), 0, stream>>>(out_eff, a_ff + 256, ff_src, ff_dst,
                                                              df0, df1, s_ff, sum_f, E_FF);
  edge_scores<<<dim3((E_RF + 7) / 8), dim3(256), 0, stream>>>(out_erf, a_rf + 256, rf_src, rf_dst,
                                                              dr2, df2, s_rf, sum_f, E_RF);
  edge_scores<<<dim3((E_RR + 7) / 8), dim3(256), 0, stream>>>(out_err, a_rr + 256, rr_src, rr_dst,
                                                              dr0, dr1, s_rr, sum_r, E_RR);

  // 6) edge updates (normalize by SRC node's sum, as in reference) + dst aggregation
  edge_update<<<dim3((E_FF + 7) / 8), dim3(256), 0, stream>>>(out_eff, hfc_ff, ff_src, ff_dst,
                                                              s_ff, sum_f, agg_ff, E_FF);
  edge_update<<<dim3((E_RR + 7) / 8), dim3(256), 0, stream>>>(out_err, hrc_rr, rr_src, rr_dst,
                                                              s_rr, sum_r, agg_rr, E_RR);
  edge_update<<<dim3((E_RF + 7) / 8), dim3(256), 0, stream>>>(out_erf, hfc_rf, rf_src, rf_dst,
                                                              s_rf, sum_r, agg_rf, E_RF);

  // 7) node updates: new = h + lrelu(agg @ Wnc [+ agg2 @ Wnc2]), fused epilogue
  node_update_wmma<<<dim3(TILES(N_F), D_NODE / 16), wave, 0, stream>>>(
      agg_ff, Wnc_ff, agg_rf, Wnc_rf, hf, out_f, N_F, D_NODE, D_EDGE);
  node_update_wmma<<<dim3(TILES(N_R), D_NODE / 16), wave, 0, stream>>>(
      agg_rr, Wnc_rr, nullptr, nullptr, hr, out_r, N_R, D_NODE, D_EDGE);

  #undef TILES
}